// CNN_LeNetStd_79714593013874
// MI455X (gfx1250) — compile-verified
//
#include <hip/hip_runtime.h>
#include <math.h>

typedef float v2f __attribute__((ext_vector_type(2)));
typedef float v8f __attribute__((ext_vector_type(8)));
typedef unsigned int u32x4 __attribute__((ext_vector_type(4)));
typedef int i32x4 __attribute__((ext_vector_type(4)));
typedef int i32x8 __attribute__((ext_vector_type(8)));

// ---------------------------------------------------------------------------
// TDM: 1-D tensor load global -> LDS (contiguous, 4-byte elements).
// D# per CDNA5 ISA ch.10.8: group0 = {count=1, lds_addr, global_addr, type=2},
// group1 = {data_size=4B, tensor_dim0=n, tensor_dim1=1, tile_dim0=n,
// tile_dim1=1}. Groups 2/3 zero (tensor <= 2D). Tracked by TENSORcnt.
// Issue from ONE wave; publish to the block with s_wait_tensorcnt + barrier.
// ---------------------------------------------------------------------------
__device__ __forceinline__ void tdm_load_1d(unsigned int lds_byte_addr,
                                            const void* gptr,
                                            unsigned int nelem) {
    const unsigned long long ga = (unsigned long long)(uintptr_t)gptr;
    u32x4 g0;
    g0[0] = 1u;                                          // count=1, user D#
    g0[1] = lds_byte_addr;                               // lds_addr
    g0[2] = (unsigned int)ga;                            // global_addr[31:0]
    g0[3] = ((unsigned int)(ga >> 32) & 0x01FFFFFFu)     // global_addr[56:32]
            | 0x80000000u;                               // type=2 ("image")
    i32x8 g1;
    g1[0] = (int)(2u << 16);                             // data_size = 4 bytes
    g1[1] = (int)((nelem & 0xFFFFu) << 16);              // tensor_dim0 lo16
    g1[2] = (int)((nelem >> 16) | (1u << 16));           // dim0 hi | tensor_dim1=1
    g1[3] = (int)((nelem & 0xFFFFu) << 16);              // tile_dim0 (= whole dim)
    g1[4] = 1;                                           // tile_dim1 = 1
    g1[5] = (int)nelem;                                  // dim0 stride (unused, 1 line)
    g1[6] = 0;
    g1[7] = 0;
    const i32x4 z4 = {0, 0, 0, 0};
#if defined(__clang_major__) && __clang_major__ >= 23
    const i32x8 z8 = {0, 0, 0, 0, 0, 0, 0, 0};
    __builtin_amdgcn_tensor_load_to_lds(g0, g1, z4, z4, z8, 0);
#else
    __builtin_amdgcn_tensor_load_to_lds(g0, g1, z4, z4, 0);
#endif
}

__device__ __forceinline__ unsigned int lds_addr_of(const void* shared_ptr) {
    // Generic shared pointer: ADDR[31:0] is the LDS byte address (aperture map).
    return (unsigned int)(uintptr_t)shared_ptr;
}

// ---------------------------------------------------------------------------
// Kernel 1: conv1 (1->6, 5x5, VALID) + relu + 2x2 maxpool, one image / block.
// x: [B,1,32,32] -> h1: [B,6,14,14]
// ---------------------------------------------------------------------------
__global__ void __launch_bounds__(256)
conv1_relu_pool(const float* __restrict__ x, const float* __restrict__ w1,
                float* __restrict__ h1) {
    __shared__ float xs[1024];   // 32x32 image
    __shared__ float ws[160];    // 6*25 weights
    const int b   = blockIdx.x;
    const int tid = threadIdx.x;
    // image is 256 float4s: one per thread, fully coalesced
    ((float4*)xs)[tid] = ((const float4*)(x + (size_t)b * 1024))[tid];
    if (tid < 150) ws[tid] = w1[tid];
    __syncthreads();
    for (int i = tid; i < 1176; i += 256) {            // 6*14*14 pooled outputs
        const int c   = i / 196;
        const int rem = i - c * 196;
        const int py  = rem / 14, px = rem - py * 14;
        const float* wc = &ws[c * 25];
        float mx = -INFINITY;
        #pragma unroll
        for (int dy = 0; dy < 2; ++dy) {
            #pragma unroll
            for (int dx = 0; dx < 2; ++dx) {
                const int oy = 2 * py + dy, ox = 2 * px + dx;
                float s = 0.f;
                #pragma unroll
                for (int ky = 0; ky < 5; ++ky)
                    #pragma unroll
                    for (int kx = 0; kx < 5; ++kx)
                        s = fmaf(xs[(oy + ky) * 32 + ox + kx], wc[ky * 5 + kx], s);
                mx = fmaxf(mx, fmaxf(s, 0.f));
            }
        }
        h1[b * 1176 + i] = mx;
    }
}

// ---------------------------------------------------------------------------
// Kernel 2: conv2 (6->16, 5x5) as implicit GEMM via V_WMMA_F32_16X16X4_F32,
// fused relu + 2x2 maxpool, one image / block (7 waves = 224 threads).
// M = 100 output pixels (padded 112), N = 16 channels, K = 150 (pad 152).
// Input image staged by the Tensor Data Mover; branchless WMMA inner loop
// (im2col offsets from an LDS table; K-tail neutralized by zero B rows;
// invalid M-rows produce garbage the pooling stage never reads).
// h1: [B,6,14,14] -> p2: [B,400] (channel-major flatten, torch .view order)
// ---------------------------------------------------------------------------
__global__ void __launch_bounds__(224)
conv2_wmma_pool(const float* __restrict__ h1, const float* __restrict__ w2,
                float* __restrict__ p2) {
    __shared__ float img[1216];      // 6*14*14 input (+pad so p+q stays in-range)
    __shared__ float wl2[152 * 16];  // weights, pair-interleaved [(k/2)][n][k&1]
    __shared__ int   qs[152];        // im2col offset table, qs[150]=qs[151]=0
    __shared__ float buf[112 * 16];  // relu'd conv output tile [pixel][channel]
    const int b   = blockIdx.x;
    const int tid = threadIdx.x;
    if (tid < 32)                    // wave 0: DMA the input image into LDS
        tdm_load_1d(lds_addr_of(img), h1 + (size_t)b * 1176, 1176);
    for (int i = tid; i < 152 * 16; i += 224) {
        const int n = i & 15, kk = i >> 4;
        // pairs (k, k+1) adjacent per column -> one ds_load_b64 per B fragment
        wl2[(kk >> 1) * 32 + n * 2 + (kk & 1)] =
            (kk < 150) ? w2[n * 150 + kk] : 0.f;     // w2 is [16][150]
    }
    if (tid >= 32 && tid < 32 + 152) {
        const int t = tid - 32;
        int q = 0;
        if (t < 150) {
            const int ic = t / 25, r = t - ic * 25;
            const int ky = r / 5, kx = r - ky * 5;
            q = ic * 196 + ky * 14 + kx;
        }
        qs[t] = q;
    }
    if (tid < 32) __builtin_amdgcn_s_wait_tensorcnt(0);
    __syncthreads();

    const int lane = tid & 31;
    const int wv   = tid >> 5;          // wave id 0..6 -> M-tile
    const int ln   = lane & 15;
    const int hi   = lane >> 4;         // half-wave select
    const int koff = hi * 2;            // K offset per WMMA f32 A/B layout
    const int row  = wv * 16 + ln;      // output pixel index (0..111)
    const int oy   = row / 10, ox = row - oy * 10;
    const int p    = oy * 14 + ox;      // lane-constant part of im2col address

    v8f c = {};
    #pragma unroll 2
    for (int k = 0; k < 152; k += 4) {
        const int k0 = k + koff;                      // even
        typedef int v2i __attribute__((ext_vector_type(2)));
        const v2i q2 = *(const v2i*)&qs[k0];          // ds_load_b64
        v2f a, bm;
        a.x = img[p + q2.x];
        a.y = img[p + q2.y];
        bm  = *(const v2f*)&wl2[(k0 >> 1) * 32 + ln * 2];  // ds_load_b64
        c = __builtin_amdgcn_wmma_f32_16x16x4_f32(false, a, false, bm,
                                                  (short)0, c, false, false);
    }
    // C layout: VGPR r, lanes 0-15 -> M=r, lanes 16-31 -> M=8+r ; col = ln
    #pragma unroll
    for (int r = 0; r < 8; ++r) {
        const int m = wv * 16 + r + hi * 8;
        buf[m * 16 + ln] = fmaxf(c[r], 0.f);
    }
    __syncthreads();
    // 2x2 maxpool over the 10x10 map; write [c*25 + py*5 + px] flatten order
    for (int i = tid; i < 400; i += 224) {
        const int ch  = i / 25;
        const int rem = i - ch * 25;
        const int py  = rem / 5, px = rem - py * 5;
        const int base = (2 * py * 10 + 2 * px) * 16 + ch;
        const float m0 = fmaxf(buf[base],       buf[base + 16]);   // ox, ox+1
        const float m1 = fmaxf(buf[base + 160], buf[base + 176]);  // oy+1 row
        p2[b * 400 + i] = fmaxf(m0, m1);
    }
}

// ---------------------------------------------------------------------------
// Kernel 3: O = relu(A @ W^T), A:[M,K], W:[N,K], O:[M,N]. K,N compile-time.
// Block = one 16-row M-tile; the contiguous 16xK A-tile is staged into LDS by
// the Tensor Data Mover (wave 0 issues, TENSORcnt + barrier publishes).
// One wave per 16-col N-tile, WMMA f32 16x16x4 over K.
// fc1: K=400,N=120,NT=8 ; fc2: K=120,N=84,NT=6.
// ---------------------------------------------------------------------------
template <int K, int N, int NT>
__global__ void __launch_bounds__(32 * NT)
fc_wmma_relu(const float* __restrict__ A, const float* __restrict__ W,
             float* __restrict__ O) {
    __shared__ float Alds[16 * K];
    const int tid  = threadIdx.x;
    const int mt   = blockIdx.x;
    const int lane = tid & 31;
    const int nt   = tid >> 5;
    const int ln   = lane & 15;
    const int hi   = lane >> 4;
    const int koff = hi * 2;
    if (tid < 32) {   // wave 0: DMA the contiguous 16*K-float tile into LDS
        tdm_load_1d(lds_addr_of(Alds), A + (size_t)mt * 16 * K, 16 * K);
        __builtin_amdgcn_s_wait_tensorcnt(0);
    }
    __syncthreads();

    const int  n      = nt * 16 + ln;
    const bool nvalid = n < N;
    const int  nw     = nvalid ? n : 0;        // clamped: garbage cols never stored
    const float* Arow = &Alds[ln * K];
    const float* Wrow = &W[(size_t)nw * K];

    v8f c = {};
    #pragma unroll 4
    for (int k = 0; k < K; k += 4) {
        const v2f a  = *(const v2f*)&Arow[k + koff];   // ds_load_b64
        const v2f bm = *(const v2f*)&Wrow[k + koff];   // global_load_b64 (L2-hot)
        c = __builtin_amdgcn_wmma_f32_16x16x4_f32(false, a, false, bm,
                                                  (short)0, c, false, false);
    }
    if (nvalid) {
        #pragma unroll
        for (int r = 0; r < 8; ++r) {
            const int m = mt * 16 + r + hi * 8;
            O[(size_t)m * N + n] = fmaxf(c[r], 0.f);
        }
    }
}

// ---------------------------------------------------------------------------
// Kernel 4: fc3 (84 -> 10, with bias) + softmax, one thread per row.
// ---------------------------------------------------------------------------
__global__ void __launch_bounds__(256)
fc3_softmax(const float* __restrict__ A, const float* __restrict__ W,
            const float* __restrict__ bias, float* __restrict__ out, int M) {
    const int row = blockIdx.x * 256 + threadIdx.x;
    if (row >= M) return;
    float xv[84];
    #pragma unroll
    for (int i = 0; i < 84; ++i) xv[i] = A[(size_t)row * 84 + i];
    float lg[10];
    float mx = -INFINITY;
    #pragma unroll
    for (int j = 0; j < 10; ++j) {
        float s = bias[j];
        #pragma unroll
        for (int i = 0; i < 84; ++i) s = fmaf(xv[i], W[j * 84 + i], s);
        lg[j] = s;
        mx = fmaxf(mx, s);
    }
    float den = 0.f;
    #pragma unroll
    for (int j = 0; j < 10; ++j) { lg[j] = __expf(lg[j] - mx); den += lg[j]; }
    const float inv = 1.f / den;
    #pragma unroll
    for (int j = 0; j < 10; ++j) out[(size_t)row * 10 + j] = lg[j] * inv;
}

// ---------------------------------------------------------------------------
extern "C" void kernel_launch(void* const* d_in, const int* in_sizes, int n_in,
                              void* d_out, int out_size, void* d_ws, size_t ws_size,
                              hipStream_t stream) {
    const float* x    = (const float*)d_in[0];   // [B,1,32,32]
    const float* w1   = (const float*)d_in[1];   // [6,1,5,5]
    const float* w2   = (const float*)d_in[2];   // [16,6,5,5]
    const float* fw1  = (const float*)d_in[3];   // [120,400]
    const float* fw2  = (const float*)d_in[4];   // [84,120]
    const float* fc3w = (const float*)d_in[5];   // [10,84]
    const float* fc3b = (const float*)d_in[6];   // [10]
    float* out = (float*)d_out;

    const int B = in_sizes[0] / 1024;            // 16384
    float* ws = (float*)d_ws;
    // Workspace layout (floats). a1/a2 reuse h1's region once it is dead.
    float* h1 = ws;                              // B*1176
    float* p2 = ws + (size_t)B * 1176;           // B*400
    float* a1 = ws;                              // B*120 (aliases dead h1)
    float* a2 = ws + (size_t)B * 120;            // B*84  (still inside old h1)

    conv1_relu_pool<<<B, 256, 0, stream>>>(x, w1, h1);
    conv2_wmma_pool<<<B, 224, 0, stream>>>(h1, w2, p2);
    fc_wmma_relu<400, 120, 8><<<B / 16, 256, 0, stream>>>(p2, fw1, a1);
    fc_wmma_relu<120,  84, 6><<<B / 16, 192, 0, stream>>>(a1, fw2, a2);
    fc3_softmax<<<(B + 255) / 256, 256, 0, stream>>>(a2, fc3w, fc3b, out, B);
}